// MmModel_18992345382987
// MI455X (gfx1250) — compile-verified
//
#include <hip/hip_runtime.h>
#include <hip/hip_bf16.h>
#include <stdint.h>

// ---------------- problem constants (from reference) ----------------
#define N_USERS   100000
#define N_ITEMS   50000
#define NNODES    150000          // N_USERS + N_ITEMS
#define EMBED     128
#define NLAYERS   3
#define BATCH     16384
#define EDGE_TILE 32              // edges staged per wave per async round

// ---------------- CDNA5 async-to-LDS availability ----------------
#if defined(__gfx1250__) && __has_builtin(__builtin_amdgcn_global_load_async_to_lds_b64)
#define HAVE_ASYNC_LDS 1
#else
#define HAVE_ASYNC_LDS 0
#endif

typedef int v2i __attribute__((vector_size(2 * sizeof(int))));
typedef __attribute__((address_space(1))) v2i* gptr_v2i;
typedef __attribute__((address_space(3))) v2i* lptr_v2i;

__device__ __forceinline__ void wait_async0() {
#if defined(__gfx1250__)
#if __has_builtin(__builtin_amdgcn_s_wait_asynccnt)
  __builtin_amdgcn_s_wait_asynccnt(0);
#else
  asm volatile("s_wait_asynccnt 0" ::: "memory");
#endif
#endif
}

// ---------------- hot kernel first (so the disasm snippet shows it) ----------------
// One wave32 per destination row: y[row] = sum_e w_e * x[col_e]; acc[row] += y[row].
// Interleaved (col, val_bits) CSR edges staged into LDS via one async-to-LDS b64
// DMA per lane (32 lanes x 8B = one 32-edge tile per wave per instruction).
__global__ void __launch_bounds__(256)
spmm_row_kernel(const float* __restrict__ x, const int2* __restrict__ edges,
                const int* __restrict__ rowStart,
                float* __restrict__ y, float* __restrict__ acc) {
  __shared__ __align__(16) int2 s_edge[8][EDGE_TILE];
  int wave = threadIdx.x >> 5;
  int lane = threadIdx.x & 31;
  int row  = blockIdx.x * 8 + wave;
  if (row >= NNODES) return;

  int beg = rowStart[row];
  int end = rowStart[row + 1];
  float4 sum = make_float4(0.f, 0.f, 0.f, 0.f);

  for (int t = beg; t < end; t += EDGE_TILE) {
    int cnt = end - t;
    if (cnt > EDGE_TILE) cnt = EDGE_TILE;
#if HAVE_ASYNC_LDS
    // Each lane DMAs one 8B edge record; edge array is padded by EDGE_TILE so
    // the unconditional full-tile copy is always in-bounds. Over-fetched
    // entries beyond `cnt` are never consumed.
    __builtin_amdgcn_global_load_async_to_lds_b64(
        (gptr_v2i)(edges + t + lane),
        (lptr_v2i)(&s_edge[wave][lane]), 0, 0);
    wait_async0();
#else
    if (lane < cnt) s_edge[wave][lane] = edges[t + lane];
#endif
    for (int j = 0; j < cnt; ++j) {
      int2  e = s_edge[wave][j];
      int   c = e.x;
      float w = __int_as_float(e.y);
      float4 xv = *reinterpret_cast<const float4*>(x + (size_t)c * EMBED + lane * 4);
      sum.x = fmaf(w, xv.x, sum.x);
      sum.y = fmaf(w, xv.y, sum.y);
      sum.z = fmaf(w, xv.z, sum.z);
      sum.w = fmaf(w, xv.w, sum.w);
    }
  }

  size_t base = (size_t)row * EMBED + lane * 4;
  *reinterpret_cast<float4*>(y + base) = sum;         // next-layer embedding
  float4 a = *reinterpret_cast<const float4*>(acc + base);
  a.x += sum.x; a.y += sum.y; a.z += sum.z; a.w += sum.w;
  *reinterpret_cast<float4*>(acc + base) = a;         // fused acc += e
}

// acc = E0 ; e_cur = E0  (float4 streaming copy)
__global__ void __launch_bounds__(256)
init_kernel(const float4* __restrict__ E0, float4* __restrict__ acc,
            float4* __restrict__ ecur, int n4) {
  int i = blockIdx.x * blockDim.x + threadIdx.x;
  if (i < n4) {
    float4 v = E0[i];
    acc[i]  = v;
    ecur[i] = v;
  }
}

__global__ void __launch_bounds__(256)
zero_int2_kernel(int* __restrict__ a, int* __restrict__ b, int n) {
  int i = blockIdx.x * blockDim.x + threadIdx.x;
  if (i < n) { a[i] = 0; b[i] = 0; }
}

// deg[r]++ for every edge (rows streamed nontemporally: one-shot read)
__global__ void __launch_bounds__(256)
count_kernel(const int* __restrict__ rows, int* __restrict__ deg, int nnz) {
  int k = blockIdx.x * blockDim.x + threadIdx.x;
  if (k < nnz) {
    int r = __builtin_nontemporal_load(rows + k);
    atomicAdd(&deg[r], 1);
  }
}

// exclusive scan of deg -> rowStart (single 1024-thread block, Hillis-Steele per chunk)
__global__ void __launch_bounds__(1024)
scan_kernel(const int* __restrict__ deg, int* __restrict__ rowStart, int n) {
  __shared__ int tmp[1024];
  __shared__ int running;
  int tid = threadIdx.x;
  if (tid == 0) running = 0;
  __syncthreads();
  for (int base = 0; base < n; base += 1024) {
    int i = base + tid;
    int v = (i < n) ? deg[i] : 0;
    tmp[tid] = v;
    __syncthreads();
    for (int off = 1; off < 1024; off <<= 1) {
      int t = (tid >= off) ? tmp[tid - off] : 0;
      __syncthreads();
      tmp[tid] += t;
      __syncthreads();
    }
    int incl = tmp[tid];
    if (i < n) rowStart[i] = running + (incl - v);
    __syncthreads();
    if (tid == 0) running += tmp[1023];
    __syncthreads();
  }
  if (tid == 0) rowStart[n] = running;
}

// scatter edges into row-sorted interleaved CSR: edge[d] = (col, bits(val))
__global__ void __launch_bounds__(256)
scatter_kernel(const int* __restrict__ rows, const int* __restrict__ cols,
               const float* __restrict__ vals, const int* __restrict__ rowStart,
               int* __restrict__ cursor, int2* __restrict__ edges, int nnz) {
  int k = blockIdx.x * blockDim.x + threadIdx.x;
  if (k >= nnz) return;
  int   r = __builtin_nontemporal_load(rows + k);
  int   c = __builtin_nontemporal_load(cols + k);
  float v = __builtin_nontemporal_load(vals + k);
  int p = atomicAdd(&cursor[r], 1);
  int d = rowStart[r] + p;
  int2 e; e.x = c; e.y = __float_as_int(v);
  edges[d] = e;
}

// out rows 0..B-1:   0.25*acc[user_idx]
// out rows B..2B-1:  0.25*acc[N_USERS + pos_idx]
// out rows 2B..3B-1: 0.25*acc[N_USERS + neg_idx]
__global__ void __launch_bounds__(256)
gather_kernel(const float* __restrict__ acc, const int* __restrict__ u,
              const int* __restrict__ p, const int* __restrict__ ng,
              float* __restrict__ out) {
  int tid  = blockIdx.x * blockDim.x + threadIdx.x;
  int row  = tid >> 5;
  int lane = tid & 31;
  if (row >= 3 * BATCH) return;
  int seg = row / BATCH;
  int idx = row - seg * BATCH;
  int node = (seg == 0) ? u[idx] : (((seg == 1) ? p[idx] : ng[idx]) + N_USERS);
  float4 v = *reinterpret_cast<const float4*>(acc + (size_t)node * EMBED + lane * 4);
  v.x *= 0.25f; v.y *= 0.25f; v.z *= 0.25f; v.w *= 0.25f;
  *reinterpret_cast<float4*>(out + (size_t)row * EMBED + lane * 4) = v;
}

// ---------------- launcher ----------------
extern "C" void kernel_launch(void* const* d_in, const int* in_sizes, int n_in,
                              void* d_out, int out_size, void* d_ws, size_t ws_size,
                              hipStream_t stream) {
  const int*   uidx = (const int*)d_in[0];
  const int*   pidx = (const int*)d_in[1];
  const int*   nidx = (const int*)d_in[2];
  const float* E0   = (const float*)d_in[3];
  const int*   rows = (const int*)d_in[4];
  const int*   cols = (const int*)d_in[5];
  const float* vals = (const float*)d_in[6];
  const int nnz = in_sizes[4];   // 2 * N_EDGES

  // carve workspace (256B-aligned)
  char* w = (char*)d_ws;
  auto carve = [&](size_t bytes) -> void* {
    void* ptr = (void*)w;
    w += (bytes + 255) & ~(size_t)255;
    return ptr;
  };
  const size_t embBytes = (size_t)NNODES * EMBED * sizeof(float);
  float* acc      = (float*)carve(embBytes);
  float* eA       = (float*)carve(embBytes);
  float* eB       = (float*)carve(embBytes);
  int*   rowStart = (int*)carve((size_t)(NNODES + 1) * sizeof(int));
  int*   deg      = (int*)carve((size_t)NNODES * sizeof(int));
  int*   cursor   = (int*)carve((size_t)NNODES * sizeof(int));
  int2*  edges    = (int2*)carve(((size_t)nnz + EDGE_TILE) * sizeof(int2));

  const int B = 256;
  // 1) acc = e_cur = E0
  {
    int n4 = NNODES * (EMBED / 4);
    init_kernel<<<(n4 + B - 1) / B, B, 0, stream>>>(
        (const float4*)E0, (float4*)acc, (float4*)eA, n4);
  }
  // 2) build interleaved CSR: count -> scan -> scatter
  zero_int2_kernel<<<(NNODES + B - 1) / B, B, 0, stream>>>(deg, cursor, NNODES);
  count_kernel<<<(nnz + B - 1) / B, B, 0, stream>>>(rows, deg, nnz);
  scan_kernel<<<1, 1024, 0, stream>>>(deg, rowStart, NNODES);
  scatter_kernel<<<(nnz + B - 1) / B, B, 0, stream>>>(
      rows, cols, vals, rowStart, cursor, edges, nnz);

  // 3) three propagation layers (gather-style SpMM, fused accumulate)
  float* cur = eA;
  float* nxt = eB;
  for (int l = 0; l < NLAYERS; ++l) {
    spmm_row_kernel<<<(NNODES + 7) / 8, B, 0, stream>>>(
        cur, edges, rowStart, nxt, acc);
    float* t = cur; cur = nxt; nxt = t;
  }

  // 4) mean (x0.25) + batched gathers -> d_out
  {
    int nthreads = 3 * BATCH * 32;
    gather_kernel<<<(nthreads + B - 1) / B, B, 0, stream>>>(
        acc, uidx, pidx, nidx, (float*)d_out);
  }
}